// DatasetScoreMatchingLoss_40673340293719
// MI455X (gfx1250) — compile-verified
//
#include <hip/hip_runtime.h>

#define NUM_GROUPS 16
#define MIN_COUNT  10.0f

typedef __attribute__((ext_vector_type(2))) float v2f;
typedef __attribute__((ext_vector_type(8))) float v8f;

// ---------------------------------------------------------------------------
// Workspace layout:
//   gAcc[64]  : float accumulators, slot g*4 + {0:sum_pos, 1:cnt_pos, 2:sum_neg, 3:cnt_neg}
//   bitmap[]  : 1 bit per dataset element, "overwritten by batch scatter"
// ---------------------------------------------------------------------------

__global__ void dsml_zero_kernel(unsigned* __restrict__ ws, int words) {
    int i = blockIdx.x * blockDim.x + threadIdx.x;
    int stride = gridDim.x * blockDim.x;
    for (; i < words; i += stride) ws[i] = 0u;
}

__device__ __forceinline__ void dsml_global_contrib(float* gAcc, float s, int lab, int g, float sign) {
    // valid = !isnan(score) && lab>=0 && g>=0 ; segment ids >= NUM_GROUPS are dropped
    if (!((s == s) && lab >= 0 && g >= 0 && g < NUM_GROUPS)) return;
    if (lab == 1) {
        unsafeAtomicAdd(&gAcc[g * 4 + 0], sign * s);
        unsafeAtomicAdd(&gAcc[g * 4 + 1], sign);
    } else if (lab == 0) {
        unsafeAtomicAdd(&gAcc[g * 4 + 2], sign * s);
        unsafeAtomicAdd(&gAcc[g * 4 + 3], sign);
    }
}

// Batch overlay: claim index (duplicate indices -> one winner, like JAX .set),
// add new contribution, subtract the buffer value it overwrites. The big
// streaming kernel can then read the original buffers unmodified.
__global__ void dsml_scatter_kernel(const float* __restrict__ probs,
                                    const int*   __restrict__ labels,
                                    const int*   __restrict__ groups,
                                    const int*   __restrict__ indices,
                                    const float* __restrict__ sbuf,
                                    const int*   __restrict__ lbuf,
                                    const int*   __restrict__ gbuf,
                                    float* __restrict__ gAcc,
                                    unsigned* __restrict__ bitmap,
                                    int B) {
    int j = blockIdx.x * blockDim.x + threadIdx.x;
    if (j >= B) return;
    int idx = indices[j];
    unsigned bit = 1u << (idx & 31);
    unsigned old = atomicOr(&bitmap[idx >> 5], bit);
    if (old & bit) return;  // another batch element already owns this slot
    dsml_global_contrib(gAcc, probs[j], labels[j], groups[j], 1.0f);
    dsml_global_contrib(gAcc, sbuf[idx], lbuf[idx], gbuf[idx], -1.0f);
}

// ---------------------------------------------------------------------------
// Main streaming reduction: 192 MB @ 23.3 TB/s ≈ 8.2 us roofline.
// LDS-privatized histograms: 64 copies, stride 65 dwords -> within a wave,
// bank = (copy + slot) % 64 is distinct per lane => conflict-free ds_add_f32.
// ---------------------------------------------------------------------------
#define HIST_COPIES 64
#define HIST_STRIDE 65

__device__ __forceinline__ void dsml_elem(float* my, float s, int lab, int g) {
    if (!((s == s) && lab >= 0 && g >= 0 && g < NUM_GROUPS)) return;
    int base;
    if (lab == 1)      base = g * 4 + 0;
    else if (lab == 0) base = g * 4 + 2;
    else return;
    __hip_atomic_fetch_add(&my[base],     s,    __ATOMIC_RELAXED, __HIP_MEMORY_SCOPE_WORKGROUP);
    __hip_atomic_fetch_add(&my[base + 1], 1.0f, __ATOMIC_RELAXED, __HIP_MEMORY_SCOPE_WORKGROUP);
}

__global__ __launch_bounds__(256) void dsml_stream_kernel(const float* __restrict__ sbuf,
                                                          const int*   __restrict__ lbuf,
                                                          const int*   __restrict__ gbuf,
                                                          float* __restrict__ gAcc,
                                                          int N) {
    __shared__ float hist[HIST_COPIES * HIST_STRIDE];
    for (int i = threadIdx.x; i < HIST_COPIES * HIST_STRIDE; i += 256) hist[i] = 0.0f;
    __syncthreads();

    float* my = &hist[(threadIdx.x & (HIST_COPIES - 1)) * HIST_STRIDE];

    int gid    = blockIdx.x * 256 + threadIdx.x;
    int stride = gridDim.x * 256;
    int n4     = N >> 2;

    const float4* s4 = (const float4*)sbuf;
    const int4*   l4 = (const int4*)lbuf;
    const int4*   g4 = (const int4*)gbuf;

    for (int i = gid; i < n4; i += stride) {
        // gfx1250 global_prefetch_b8 for the next grid-stride tile
        __builtin_prefetch(&s4[i + stride], 0, 1);
        __builtin_prefetch(&l4[i + stride], 0, 1);
        __builtin_prefetch(&g4[i + stride], 0, 1);
        float4 s = s4[i];
        int4   lb = l4[i];
        int4   gp = g4[i];
        dsml_elem(my, s.x, lb.x, gp.x);
        dsml_elem(my, s.y, lb.y, gp.y);
        dsml_elem(my, s.z, lb.z, gp.z);
        dsml_elem(my, s.w, lb.w, gp.w);
    }
    // scalar tail
    int rem = N - (n4 << 2);
    if (gid < rem) {
        int i = (n4 << 2) + gid;
        dsml_elem(my, sbuf[i], lbuf[i], gbuf[i]);
    }

    __syncthreads();
    if (threadIdx.x < 64) {
        float sum = 0.0f;
        #pragma unroll 8
        for (int c = 0; c < HIST_COPIES; ++c) sum += hist[c * HIST_STRIDE + threadIdx.x];
        unsafeAtomicAdd(&gAcc[threadIdx.x], sum);
    }
}

// ---------------------------------------------------------------------------
// Finalize: cross-group reductions via V_WMMA_F32_16X16X4_F32 (f32-exact).
// A = ones(16x4); B columns carry per-group stats; chain 4 WMMAs (K=4 each)
// to sum over all 16 groups. Two passes: (n, sum_avg) then sum incl*(avg-mean)^2
// -- the two-pass form avoids E[x^2]-mean^2 cancellation.
// Assumed B 4x16 f32 layout (rows striped like A's K striping):
//   VGPR0: lanes0-15 = row0 cols, lanes16-31 = row1 cols
//   VGPR1: lanes0-15 = row2 cols, lanes16-31 = row3 cols
// D row M=0 sits in C-VGPR0 lanes 0-15 (N = lane).
// ---------------------------------------------------------------------------
__device__ __forceinline__ void dsml_group_stats(const float* gAcc, int g, int pos,
                                                 float& avg, float& incl) {
    float sum = gAcc[g * 4 + (pos ? 0 : 2)];
    float cnt = gAcc[g * 4 + (pos ? 1 : 3)];
    avg  = sum / fmaxf(cnt, 1.0f);
    incl = (cnt >= MIN_COUNT) ? 1.0f : 0.0f;
}

__global__ __launch_bounds__(32) void dsml_finalize_kernel(const float* __restrict__ gAcc,
                                                           float* __restrict__ out) {
    int lane = threadIdx.x;        // exactly one wave32, EXEC all ones for WMMA
    int h = lane >> 4;             // which half of the wave
    int l = lane & 15;             // column index n

    v2f a; a.x = 1.0f; a.y = 1.0f; // ones A matrix (all M rows, all K)

    // ---- pass 1: columns {0:n_pos, 1:s1_pos, 2:n_neg, 3:s1_neg} ----
    v8f c1 = {};
    for (int t = 0; t < 4; ++t) {
        int g0 = 4 * t + h;        // row h   (VGPR0)
        int g1 = 4 * t + 2 + h;    // row 2+h (VGPR1)
        float av, ic;
        v2f b; b.x = 0.0f; b.y = 0.0f;
        if (l < 4) {
            int pos = (l < 2);
            dsml_group_stats(gAcc, g0, pos, av, ic);
            b.x = (l & 1) ? ic * av : ic;
            dsml_group_stats(gAcc, g1, pos, av, ic);
            b.y = (l & 1) ? ic * av : ic;
        }
        c1 = __builtin_amdgcn_wmma_f32_16x16x4_f32(false, a, false, b, (short)0, c1, false, false);
    }
    float r1 = c1[0];
    float np  = __shfl(r1, 0, 32);
    float s1p = __shfl(r1, 1, 32);
    float nn  = __shfl(r1, 2, 32);
    float s1n = __shfl(r1, 3, 32);
    float meanp = s1p / fmaxf(np, 1.0f);
    float meann = s1n / fmaxf(nn, 1.0f);

    // ---- pass 2: columns {0: sum incl*(avg-mean_p)^2 (pos), 1: same (neg)} ----
    v8f c2 = {};
    for (int t = 0; t < 4; ++t) {
        int g0 = 4 * t + h;
        int g1 = 4 * t + 2 + h;
        float av, ic;
        v2f b; b.x = 0.0f; b.y = 0.0f;
        if (l < 2) {
            int pos = (l == 0);
            float mean = pos ? meanp : meann;
            dsml_group_stats(gAcc, g0, pos, av, ic);
            float d0 = av - mean; b.x = ic * d0 * d0;
            dsml_group_stats(gAcc, g1, pos, av, ic);
            float d1 = av - mean; b.y = ic * d1 * d1;
        }
        c2 = __builtin_amdgcn_wmma_f32_16x16x4_f32(false, a, false, b, (short)0, c2, false, false);
    }
    float r2 = c2[0];
    float vps = __shfl(r2, 0, 32);
    float vns = __shfl(r2, 1, 32);

    if (lane == 0) {
        float varp = vps / fmaxf(np - 1.0f, 1.0f);
        float varn = vns / fmaxf(nn - 1.0f, 1.0f);
        bool pos_ok = (np >= 2.0f);
        bool neg_ok = (nn >= 2.0f);
        float loss;
        if (pos_ok && neg_ok)      loss = 0.5f * (varp + varn);
        else if (pos_ok)           loss = varp;
        else if (neg_ok)           loss = varn;
        else                       loss = 0.0f;
        out[0] = loss;
    }
}

// ---------------------------------------------------------------------------
extern "C" void kernel_launch(void* const* d_in, const int* in_sizes, int n_in,
                              void* d_out, int out_size, void* d_ws, size_t ws_size,
                              hipStream_t stream) {
    const float* probs   = (const float*)d_in[0];
    const int*   labels  = (const int*)d_in[1];
    const int*   groups  = (const int*)d_in[2];
    const int*   indices = (const int*)d_in[3];
    const float* sbuf    = (const float*)d_in[4];
    const int*   lbuf    = (const int*)d_in[5];
    const int*   gbuf    = (const int*)d_in[6];

    int B = in_sizes[0];
    int N = in_sizes[4];

    float*    gAcc   = (float*)d_ws;
    unsigned* bitmap = (unsigned*)d_ws + 64;
    int bitmapWords  = (N + 31) / 32;
    int zeroWords    = 64 + bitmapWords;

    dsml_zero_kernel<<<1024, 256, 0, stream>>>((unsigned*)d_ws, zeroWords);
    dsml_scatter_kernel<<<(B + 255) / 256, 256, 0, stream>>>(
        probs, labels, groups, indices, sbuf, lbuf, gbuf, gAcc, bitmap, B);
    dsml_stream_kernel<<<2048, 256, 0, stream>>>(sbuf, lbuf, gbuf, gAcc, N);
    dsml_finalize_kernel<<<1, 32, 0, stream>>>(gAcc, (float*)d_out);
}